// MiniMindMOEFeedForward_11106785427919
// MI455X (gfx1250) — compile-verified
//
#include <hip/hip_runtime.h>
#include <math.h>

// MoE forward (top-2 of 8 experts + shared expert), f32, for MI455X/gfx1250.
// Grouped-GEMM over compacted per-expert token lists on the f32 matrix path
// (V_WMMA_F32_16X16X4_F32), with double-buffered GLOBAL_LOAD_ASYNC_TO_LDS
// staging of the activation tiles (ASYNCcnt-tracked) when available.

#define H_   512
#define I_   1408
#define E_   8
#define KEXP 2
#define BSZ  2
#define SSZ  1024
#define T_   (BSZ * SSZ)
#define ALPHA_ 0.1f
#define TM   32   // rows (tokens) per block tile

typedef __attribute__((ext_vector_type(2))) float v2f;
typedef __attribute__((ext_vector_type(8))) float v8f;

#if defined(__has_builtin)
#if __has_builtin(__builtin_amdgcn_global_load_async_to_lds_b32) && \
    __has_builtin(__builtin_amdgcn_s_wait_asynccnt)
#define HAS_ASYNC_LDS 1
#endif
#endif

typedef __attribute__((address_space(1))) int gint_t;
typedef __attribute__((address_space(3))) int lint_t;

__device__ __forceinline__ v8f wmma_f32(v2f a, v2f b, v8f c) {
  // 8 args: (neg_a, A, neg_b, B, c_mod, C, reuse_a, reuse_b)
  return __builtin_amdgcn_wmma_f32_16x16x4_f32(false, a, false, b, (short)0, c,
                                               false, false);
}

// Copy one f32 element global->LDS (async global_load_async_to_lds_b32 when
// the toolchain exposes it; falls back to a plain staged copy otherwise).
__device__ __forceinline__ void ld_g2l(const float* gp, float* lp) {
#if HAS_ASYNC_LDS
  __builtin_amdgcn_global_load_async_to_lds_b32(
      (gint_t*)(void*)(const void*)gp, (lint_t*)(void*)lp, 0, 0);
#else
  *lp = *gp;
#endif
}
__device__ __forceinline__ void wait_g2l() {
#if HAS_ASYNC_LDS
  __builtin_amdgcn_s_wait_asynccnt(0);
#endif
}

// ---------------------------------------------------------------- gate ------
__global__ void zero_kernel(int* exCnt, int* cntBE) {
  int i = threadIdx.x;
  if (i < E_) exCnt[i] = 0;
  if (i < BSZ * E_) cntBE[i] = 0;
}

__global__ void gate_kernel(const float* __restrict__ x,
                            const float* __restrict__ gw, int* exCnt, int* cntBE,
                            float* __restrict__ scoreBuf, int* __restrict__ exTok,
                            float* __restrict__ exW, int* __restrict__ tokE,
                            int* __restrict__ tokS) {
  int wave = threadIdx.x >> 5;
  int lane = threadIdx.x & 31;
  int t = blockIdx.x * 8 + wave;
  if (t >= T_) return;
  const float* xt = x + (size_t)t * H_;

  float lg[E_];
#pragma unroll
  for (int e = 0; e < E_; ++e) {
    const float* ge = gw + e * H_;
    float p = 0.f;
    for (int h = lane; h < H_; h += 32) p += xt[h] * ge[h];
#pragma unroll
    for (int off = 16; off > 0; off >>= 1) p += __shfl_xor(p, off, 32);
    lg[e] = p;  // all lanes hold the full dot after xor-reduce
  }
  float m = lg[0];
#pragma unroll
  for (int e = 1; e < E_; ++e) m = fmaxf(m, lg[e]);
  float sum = 0.f, sc[E_];
#pragma unroll
  for (int e = 0; e < E_; ++e) { sc[e] = expf(lg[e] - m); sum += sc[e]; }
#pragma unroll
  for (int e = 0; e < E_; ++e) sc[e] /= sum;

  if (lane == 0) {
#pragma unroll
    for (int e = 0; e < E_; ++e) scoreBuf[(size_t)t * E_ + e] = sc[e];
    // top-2, ties -> lowest index (lax.top_k semantics)
    int e0 = 0;
#pragma unroll
    for (int e = 1; e < E_; ++e) if (sc[e] > sc[e0]) e0 = e;
    int e1 = (e0 == 0) ? 1 : 0;
#pragma unroll
    for (int e = 0; e < E_; ++e) if (e != e0 && sc[e] > sc[e1]) e1 = e;
    float w0 = sc[e0], w1 = sc[e1];
    float d = w0 + w1 + 1e-20f;
    w0 /= d; w1 /= d;
    int b = t / SSZ;
    int s0 = atomicAdd(&exCnt[e0], 1);
    exTok[e0 * T_ + s0] = t; exW[e0 * T_ + s0] = w0;
    tokE[t * 2 + 0] = e0; tokS[t * 2 + 0] = s0;
    atomicAdd(&cntBE[b * E_ + e0], 1);
    int s1 = atomicAdd(&exCnt[e1], 1);
    exTok[e1 * T_ + s1] = t; exW[e1 * T_ + s1] = w1;
    tokE[t * 2 + 1] = e1; tokS[t * 2 + 1] = s1;
    atomicAdd(&cntBE[b * E_ + e1], 1);
  }
}

__global__ void scan_kernel(const int* exCnt, int* exBase) {
  if (threadIdx.x == 0) {
    int a = 0;
    for (int e = 0; e < E_; ++e) { exBase[e] = a; a += exCnt[e]; }
  }
}

// ------------------------------------------------- gate/up GEMM + SwiGLU ----
// 32-row x 128-col tile per block (8 waves x 16 cols), two B matrices (gate &
// up) and two A fragments share fetches: 4 WMMAs per B-fragment pair.
// A tiles double-buffered in LDS via async global->LDS copies.
__global__ __launch_bounds__(256) void gu_gemm(
    const float* __restrict__ X, const float* __restrict__ Wg,
    const float* __restrict__ Wu, unsigned long long eStride,
    const int* __restrict__ countPtr, int fixedCount,
    const int* __restrict__ tokList, const int* __restrict__ basePtr,
    float* __restrict__ actOut) {
  int e = blockIdx.z;
  int Ne = countPtr ? countPtr[e] : fixedCount;
  int rowbase = blockIdx.x * TM;
  if (rowbase >= Ne) return;
  int abase = basePtr ? basePtr[e] : 0;
  const float* Bg = Wg + (size_t)e * eStride;
  const float* Bu = Wu + (size_t)e * eStride;

  int wave = threadIdx.x >> 5, lane = threadIdx.x & 31;
  int hi = lane >> 4, c16 = lane & 15;  // c16 doubles as A-row (lane&15)
  int n0 = (blockIdx.y * 8 + wave) * 16;

  __shared__ float As[2][TM * 34];  // stride 34: conflict-free b64 frag reads
  __shared__ int tokRow[TM];
  if (threadIdx.x < TM) {
    int row = rowbase + threadIdx.x;
    tokRow[threadIdx.x] = tokList ? tokList[e * T_ + ((row < Ne) ? row : 0)]
                                  : row;
  }
  __syncthreads();

  v8f aG0 = {0,0,0,0,0,0,0,0}, aG1 = {0,0,0,0,0,0,0,0};
  v8f aU0 = {0,0,0,0,0,0,0,0}, aU1 = {0,0,0,0,0,0,0,0};

  const int NC = H_ / 32;
  // issue chunk 0 fill
  for (int i = threadIdx.x; i < TM * 32; i += 256) {
    int r = i >> 5, cc = i & 31;
    ld_g2l(X + (size_t)tokRow[r] * H_ + cc, &As[0][r * 34 + cc]);
  }
  for (int c = 0; c < NC; ++c) {
    wait_g2l();
    __syncthreads();  // single barrier/chunk (refilled buffer was last read
                      // before all waves passed the previous barrier)
    if (c + 1 < NC) {
      int kk = (c + 1) * 32;
      float* dst = As[(c + 1) & 1];
      for (int i = threadIdx.x; i < TM * 32; i += 256) {
        int r = i >> 5, cc = i & 31;
        ld_g2l(X + (size_t)tokRow[r] * H_ + kk + cc, &dst[r * 34 + cc]);
      }
    }
    const float* buf = As[c & 1];
    const float* bgp = Bg + (size_t)(c * 32) * I_ + n0 + c16;
    const float* bup = Bu + (size_t)(c * 32) * I_ + n0 + c16;
#pragma unroll
    for (int k0 = 0; k0 < 32; k0 += 4) {
      int k2 = k0 + 2 * hi;
      v2f a0 = *(const v2f*)&buf[c16 * 34 + k2];
      v2f a1 = *(const v2f*)&buf[(16 + c16) * 34 + k2];
      v2f bg; bg.x = bgp[(size_t)k2 * I_]; bg.y = bgp[(size_t)(k2 + 1) * I_];
      v2f bu; bu.x = bup[(size_t)k2 * I_]; bu.y = bup[(size_t)(k2 + 1) * I_];
      aG0 = wmma_f32(a0, bg, aG0);
      aG1 = wmma_f32(a1, bg, aG1);
      aU0 = wmma_f32(a0, bu, aU0);
      aU1 = wmma_f32(a1, bu, aU1);
    }
  }
#pragma unroll
  for (int s = 0; s < 2; ++s) {
    v8f G = s ? aG1 : aG0;
    v8f U = s ? aU1 : aU0;
#pragma unroll
    for (int j = 0; j < 8; ++j) {
      int row = rowbase + s * 16 + j + hi * 8;  // C layout: vgpr j = rows j/j+8
      if (row < Ne) {
        float g = G[j], u = U[j];
        float a = (g / (1.f + expf(-g))) * u;  // silu(g)*u
        actOut[(size_t)(abase + row) * I_ + n0 + c16] = a;
      }
    }
  }
}

// ------------------------------------------------------------ down GEMM -----
__global__ __launch_bounds__(256) void down_gemm(
    const float* __restrict__ Aact, const float* __restrict__ Wd,
    unsigned long long eStride, const int* __restrict__ countPtr,
    int fixedCount, const int* __restrict__ basePtr,
    const float* __restrict__ wPtr, float* __restrict__ Yout) {
  int e = blockIdx.z;
  int Ne = countPtr ? countPtr[e] : fixedCount;
  int rowbase = blockIdx.x * TM;
  if (rowbase >= Ne) return;
  int abase = basePtr ? basePtr[e] : 0;
  const float* Bd = Wd + (size_t)e * eStride;
  const float* Arows = Aact + (size_t)(abase + rowbase) * I_;

  int wave = threadIdx.x >> 5, lane = threadIdx.x & 31;
  int hi = lane >> 4, c16 = lane & 15;
  int n0 = (blockIdx.y * 8 + wave) * 16;

  __shared__ float As[2][TM * 34];
  v8f a0c = {0,0,0,0,0,0,0,0}, a1c = {0,0,0,0,0,0,0,0};

  const int NC = I_ / 32;
  for (int i = threadIdx.x; i < TM * 32; i += 256) {
    int r = i >> 5, cc = i & 31;
    // rows >= Ne read padded (garbage) rows; results dropped in epilogue
    ld_g2l(Arows + (size_t)r * I_ + cc, &As[0][r * 34 + cc]);
  }
  for (int c = 0; c < NC; ++c) {
    wait_g2l();
    __syncthreads();
    if (c + 1 < NC) {
      int kk = (c + 1) * 32;
      float* dst = As[(c + 1) & 1];
      for (int i = threadIdx.x; i < TM * 32; i += 256) {
        int r = i >> 5, cc = i & 31;
        ld_g2l(Arows + (size_t)r * I_ + kk + cc, &dst[r * 34 + cc]);
      }
    }
    const float* buf = As[c & 1];
    const float* bp = Bd + (size_t)(c * 32) * H_ + n0 + c16;
#pragma unroll
    for (int k0 = 0; k0 < 32; k0 += 4) {
      int k2 = k0 + 2 * hi;
      v2f a0 = *(const v2f*)&buf[c16 * 34 + k2];
      v2f a1 = *(const v2f*)&buf[(16 + c16) * 34 + k2];
      v2f b; b.x = bp[(size_t)k2 * H_]; b.y = bp[(size_t)(k2 + 1) * H_];
      a0c = wmma_f32(a0, b, a0c);
      a1c = wmma_f32(a1, b, a1c);
    }
  }
#pragma unroll
  for (int s = 0; s < 2; ++s) {
    v8f A = s ? a1c : a0c;
#pragma unroll
    for (int j = 0; j < 8; ++j) {
      int row = rowbase + s * 16 + j + hi * 8;
      if (row < Ne) {
        float w = wPtr ? wPtr[e * T_ + row] : 1.0f;
        Yout[(size_t)(abase + row) * H_ + n0 + c16] = A[j] * w;
      }
    }
  }
}

// ------------------------------------------------------------- aux + out ----
__global__ void aux_reduce(const float* __restrict__ scoreBuf,
                           float* __restrict__ scoreBE) {
  __shared__ float sm[256];
  int b = blockIdx.x >> 3, e = blockIdx.x & 7;
  float s = 0.f;
  for (int i = threadIdx.x; i < SSZ; i += 256)
    s += scoreBuf[(size_t)(b * SSZ + i) * E_ + e];
  sm[threadIdx.x] = s;
  __syncthreads();
  for (int off = 128; off > 0; off >>= 1) {
    if (threadIdx.x < off) sm[threadIdx.x] += sm[threadIdx.x + off];
    __syncthreads();
  }
  if (threadIdx.x == 0) scoreBE[b * E_ + e] = sm[0];
}

__global__ void combine_kernel(const float* __restrict__ yShared,
                               const float* __restrict__ yRouted,
                               const int* __restrict__ tokE,
                               const int* __restrict__ tokS,
                               const int* __restrict__ exBase,
                               const int* __restrict__ cntBE,
                               const float* __restrict__ scoreBE,
                               float* __restrict__ out) {
  size_t idx = (size_t)blockIdx.x * 256 + threadIdx.x;
  if (idx < (size_t)T_ * H_) {
    int t = (int)(idx / H_);
    int h = (int)(idx % H_);
    float v = yShared[idx];  // fixed-order sum: shared, top1, top2
#pragma unroll
    for (int k = 0; k < KEXP; ++k) {
      int e = tokE[t * 2 + k], s = tokS[t * 2 + k];
      v += yRouted[(size_t)(exBase[e] + s) * H_ + h];
    }
    out[idx] = v;
  }
  if (blockIdx.x == 0 && threadIdx.x == 0) {
    float aux = 0.f;
    for (int b = 0; b < BSZ; ++b) {
      float acc = 0.f;
      for (int e = 0; e < E_; ++e) {
        float cnt = (float)cntBE[b * E_ + e] / (float)(SSZ * KEXP / E_);
        float ms = scoreBE[b * E_ + e] / (float)SSZ;
        acc += cnt * ms;
      }
      aux += acc;
    }
    out[(size_t)T_ * H_] = (aux / (float)BSZ) * ALPHA_;
  }
}

// ----------------------------------------------------------------- launch ---
extern "C" void kernel_launch(void* const* d_in, const int* in_sizes, int n_in,
                              void* d_out, int out_size, void* d_ws,
                              size_t ws_size, hipStream_t stream) {
  (void)in_sizes; (void)n_in; (void)out_size; (void)ws_size;
  const float* x   = (const float*)d_in[0];
  const float* gw  = (const float*)d_in[1];
  const float* Wg  = (const float*)d_in[2];
  const float* Wu  = (const float*)d_in[3];
  const float* Wd  = (const float*)d_in[4];
  const float* Wgs = (const float*)d_in[5];
  const float* Wus = (const float*)d_in[6];
  const float* Wds = (const float*)d_in[7];

  char* ws = (char*)d_ws;
  int*   exCnt    = (int*)(ws + 0);
  int*   exBase   = (int*)(ws + 256);
  int*   cntBE    = (int*)(ws + 512);
  float* scoreBE  = (float*)(ws + 768);
  float* scoreBuf = (float*)(ws + 1024);  // T*E f32
  size_t off = 1024 + (size_t)T_ * E_ * 4;
  int*   exTok = (int*)(ws + off);   off += (size_t)E_ * T_ * 4;
  float* exW   = (float*)(ws + off); off += (size_t)E_ * T_ * 4;
  int*   tokE  = (int*)(ws + off);   off += (size_t)2 * T_ * 4;
  int*   tokS  = (int*)(ws + off);   off += (size_t)2 * T_ * 4;
  off = (off + 255) & ~(size_t)255;
  float* act  = (float*)(ws + off); off += (size_t)(T_ * KEXP + TM) * I_ * 4;
  float* yR   = (float*)(ws + off); off += (size_t)(T_ * KEXP + TM) * H_ * 4;
  float* sAct = (float*)(ws + off); off += (size_t)T_ * I_ * 4;
  float* yS   = (float*)(ws + off); off += (size_t)T_ * H_ * 4;
  // total ~48 MB of workspace

  zero_kernel<<<1, 32, 0, stream>>>(exCnt, cntBE);
  gate_kernel<<<T_ / 8, 256, 0, stream>>>(x, gw, exCnt, cntBE, scoreBuf, exTok,
                                          exW, tokE, tokS);
  scan_kernel<<<1, 32, 0, stream>>>(exCnt, exBase);

  dim3 ggu(T_ / TM, I_ / 128, E_);
  gu_gemm<<<ggu, 256, 0, stream>>>(x, Wg, Wu, (unsigned long long)H_ * I_,
                                   exCnt, 0, exTok, exBase, act);
  dim3 ggs(T_ / TM, I_ / 128, 1);
  gu_gemm<<<ggs, 256, 0, stream>>>(x, Wgs, Wus, 0ull, nullptr, T_, nullptr,
                                   nullptr, sAct);

  dim3 gdr(T_ / TM, H_ / 128, E_);
  down_gemm<<<gdr, 256, 0, stream>>>(act, Wd, (unsigned long long)I_ * H_,
                                     exCnt, 0, exBase, exW, yR);
  dim3 gds(T_ / TM, H_ / 128, 1);
  down_gemm<<<gds, 256, 0, stream>>>(sAct, Wds, 0ull, nullptr, T_, nullptr,
                                     nullptr, yS);

  aux_reduce<<<BSZ * E_, 256, 0, stream>>>(scoreBuf, scoreBE);
  combine_kernel<<<(T_ * H_ + 255) / 256, 256, 0, stream>>>(
      yS, yR, tokE, tokS, exBase, cntBE, scoreBE, (float*)d_out);
}